// CollLoss_77584289235355
// MI455X (gfx1250) — compile-verified
//
#include <hip/hip_runtime.h>

typedef __attribute__((ext_vector_type(2))) float v2f;
typedef __attribute__((ext_vector_type(8))) float v8f;

#define TPB 256
#define MAXNB 1024

// Wave32 full reduction using V_WMMA_F32_16X16X4_F32.
// A (16x4 f32, 2 VGPRs/lane): lane L<16 holds A[L][0],A[L][1]; lane L>=16 holds A[L-16][2],A[L-16][3].
// Put partial in slot 0 (k=0 for lanes 0-15, k=2 for lanes 16-31), zero in slot 1.
// B = all ones (layout-invariant). D[m][n] = p[m] + p[m+16].
// Lane L<16 holds D[0..7][L] in d[0..7]; lane L>=16 holds D[8..15][L-16].
// Sum own 8 regs, then xor-shuffle across the half-wave boundary -> total in every lane.
static __device__ __forceinline__ float wave_sum32(float x) {
  v2f a; a.x = x;   a.y = 0.0f;
  v2f b; b.x = 1.0f; b.y = 1.0f;
  v8f c = {};
  v8f d = __builtin_amdgcn_wmma_f32_16x16x4_f32(
      /*neg_a=*/false, a, /*neg_b=*/false, b,
      /*c_mod=*/(short)0, c, /*reuse_a=*/false, /*reuse_b=*/false);
  float s = ((d[0] + d[1]) + (d[2] + d[3])) + ((d[4] + d[5]) + (d[6] + d[7]));
  s += __shfl_xor(s, 16, 32);
  return s;
}

// Pass 1: each thread grid-strides over objects; 12 strided b64 (x,y) loads per
// object; per-thread distance accumulation; WMMA wave reduce + LDS combine;
// one f32 block sum written to ws[blockIdx].
__global__ __launch_bounds__(TPB) void coll_partial_kernel(
    const float* __restrict__ pred,   // 12 x 2 f32
    const float* __restrict__ mot,    // num_obj x 65 x 2 f32
    const int*   __restrict__ nobj_p, // scalar
    int max_obj,
    float* __restrict__ blocksum) {
  __shared__ float s_pred[24];
  __shared__ float s_wave[TPB / 32];

  const int tid = threadIdx.x;
  if (tid < 24) s_pred[tid] = pred[tid];
  __syncthreads();

  float px[12], py[12];
#pragma unroll
  for (int t = 0; t < 12; ++t) { px[t] = s_pred[2 * t]; py[t] = s_pred[2 * t + 1]; }

  int n = *nobj_p;
  if (n > max_obj) n = max_obj;

  const int stride = gridDim.x * TPB;
  float partial = 0.0f;
  for (int obj = blockIdx.x * TPB + tid; obj < n; obj += stride) {
    // slice index t -> original timestep 5+5t -> float offset (5+5t)*2 = 10 + 10t
    const float* base = mot + (size_t)obj * 130 + 10;
    if (obj + stride < n)  // warm next object's first line (global_prefetch_b8)
      __builtin_prefetch(mot + (size_t)(obj + stride) * 130 + 10, 0, 0);
#pragma unroll
    for (int t = 0; t < 12; ++t) {
      float mx = base[10 * t];       // merged into global_load_b64 (8B aligned)
      float my = base[10 * t + 1];
      float dx = px[t] - mx;
      float dy = py[t] - my;
      partial += __builtin_sqrtf(dx * dx + dy * dy);
    }
  }

  // All TPB threads reach here (EXEC all ones for WMMA).
  float w = wave_sum32(partial);
  if ((tid & 31) == 0) s_wave[tid >> 5] = w;
  __syncthreads();
  if (tid == 0) {
    float s = 0.0f;
#pragma unroll
    for (int i = 0; i < TPB / 32; ++i) s += s_wave[i];
    blocksum[blockIdx.x] = s;
  }
}

// Pass 2: single block reduces the nb block sums deterministically.
__global__ __launch_bounds__(TPB) void coll_final_kernel(
    const float* __restrict__ blocksum, int nb, float* __restrict__ out) {
  __shared__ float s_wave[TPB / 32];
  const int tid = threadIdx.x;
  float s = 0.0f;
  for (int i = tid; i < nb; i += TPB) s += blocksum[i];
  float w = wave_sum32(s);
  if ((tid & 31) == 0) s_wave[tid >> 5] = w;
  __syncthreads();
  if (tid == 0) {
    float t = 0.0f;
#pragma unroll
    for (int i = 0; i < TPB / 32; ++i) t += s_wave[i];
    out[0] = t;
  }
}

extern "C" void kernel_launch(void* const* d_in, const int* in_sizes, int n_in,
                              void* d_out, int out_size, void* d_ws, size_t ws_size,
                              hipStream_t stream) {
  const float* pred = (const float*)d_in[0];
  const float* mot  = (const float*)d_in[1];
  const int*   nobj = (const int*)d_in[2];

  const int max_obj = in_sizes[1] / 130;  // 65 * 2 floats per object

  int nb = MAXNB;
  size_t ws_floats = ws_size / sizeof(float);
  if ((size_t)nb > ws_floats) nb = (int)ws_floats;
  if (nb < 1) nb = 1;

  float* blocksum = (float*)d_ws;
  coll_partial_kernel<<<nb, TPB, 0, stream>>>(pred, mot, nobj, max_obj, blocksum);
  coll_final_kernel<<<1, TPB, 0, stream>>>(blocksum, nb, (float*)d_out);
}